// MobileFormerBlock_74036646248785
// MI455X (gfx1250) — compile-verified
//
#include <hip/hip_runtime.h>
#include <hip/hip_bf16.h>

// ---------------------------------------------------------------------------
// MobileFormer block for MI455X (gfx1250), fp32 end-to-end.
// The two 1x1 convs (the dominant GEMMs) run on V_WMMA_F32_16X16X4_F32.
// The block is HBM-bound (~540MB traffic vs 7.4 GFLOP), so fp32 WMMA keeps
// reference precision at no throughput cost.
// ---------------------------------------------------------------------------

typedef __attribute__((ext_vector_type(2))) float v2f;
typedef __attribute__((ext_vector_type(8))) float v8f;

// Input ordering: setup_inputs() dict insertion order, params flattened in
// insertion order (nested ac1/ac2 dicts likewise).
enum {
  IN_LOCAL = 0, IN_GLOBAL,
  P_MF_Q_W, P_MF_Q_B, P_MF_UN_W, P_MF_UN_B,
  P_F_QVK_W, P_F_QVK_B, P_F_Q_W, P_F_Q_B, P_F_K_W, P_F_K_B, P_F_V_W, P_F_V_B,
  P_F_O_W, P_F_O_B, P_F_NORM_G, P_F_NORM_B,
  P_F_MLP1_W, P_F_MLP1_B, P_F_MLP2_W, P_F_MLP2_B, P_F_MLPNORM_G, P_F_MLPNORM_B,
  P_M_CONV1_W, P_M_CONV1_B, P_M_BN1_G, P_M_BN1_B, P_M_BN1_M, P_M_BN1_V,
  P_AC1_W1, P_AC1_B1, P_AC1_W2, P_AC1_B2, P_AC1_BN_G, P_AC1_BN_B, P_AC1_BN_M, P_AC1_BN_V,
  P_M_CONV2_W, P_M_CONV2_B, P_M_BN2_G, P_M_BN2_B, P_M_BN2_M, P_M_BN2_V,
  P_AC2_W1, P_AC2_B1, P_AC2_W2, P_AC2_B2, P_AC2_BN_G, P_AC2_BN_B, P_AC2_BN_M, P_AC2_BN_V,
  P_M_CONV3_W, P_M_CONV3_B, P_M_BN3_G, P_M_BN3_B, P_M_BN3_M, P_M_BN3_V,
  P_FM_KV_W, P_FM_KV_B, NPTRS
};

struct Ptrs { const float* p[NPTRS]; };

#define HW 3136          // 56*56
#define NB 32
#define CIN 64
#define CEXP 256
#define COUT 64
#define DD 192
#define MM 6

__device__ __forceinline__ float relu6f(float x) { return fminf(fmaxf(x, 0.f), 6.f); }

#define PFOR(i, nElems) for (int i = tid; i < (nElems); i += 256)

// ---------------------------------------------------------------------------
// K1: entire Former path per image + dyn-relu coefficients + Kf/Vf.
// ---------------------------------------------------------------------------
__global__ __launch_bounds__(256) void former_kernel(
    Ptrs P, float* __restrict__ scoresWS,
    float* __restrict__ A1, float* __restrict__ B1c,
    float* __restrict__ A2, float* __restrict__ B2c,
    float* __restrict__ KF, float* __restrict__ VF,
    float* __restrict__ zout)
{
  const int n = blockIdx.x, tid = threadIdx.x;
  __shared__ float sG[MM * DD];        // global feature (1152)
  __shared__ float sQm[MM * CIN];      // mobile->former queries (384)
  __shared__ float sFus[MM * CIN];     // fusion (384)
  __shared__ float sZ[MM * DD];        // z_hidden / z_out (1152)
  __shared__ float sA[MM * 576];       // qvk (3456)
  __shared__ float sQh[MM * DD], sKh[MM * DD], sVh[MM * DD];
  __shared__ float sB[MM * 384];       // mlp hidden / head scores / t1 (2304)
  __shared__ float sC[MM * DD];        // att_in (1152)
  __shared__ float sRed[256];
  __shared__ float sScal[32];

  const float* gf = P.p[IN_GLOBAL] + (size_t)n * MM * DD;
  const float* lf = P.p[IN_LOCAL] + (size_t)n * CIN * HW;
  float* scx = scoresWS + (size_t)n * MM * HW;

  PFOR(i, MM * DD) sG[i] = gf[i];
  __syncthreads();

  // q = global @ mf_q_w + b  (6x64)
  {
    const float* W = P.p[P_MF_Q_W]; const float* b = P.p[P_MF_Q_B];
    PFOR(i, MM * CIN) {
      int m = i >> 6, c = i & 63; float a = b[c];
      for (int d = 0; d < DD; d++) a += sG[m * DD + d] * W[d * CIN + c];
      sQm[i] = a;
    }
  }
  __syncthreads();

  // scores over l, track max
  float lmax[MM];
  for (int m = 0; m < MM; m++) lmax[m] = -1e30f;
  const float invs = 1.0f / (8.0f + 1e-10f);
  for (int l = tid; l < HW; l += 256) {
    float acc[MM] = {0, 0, 0, 0, 0, 0};
    for (int c = 0; c < CIN; c++) {
      float v = lf[(size_t)c * HW + l];
      for (int m = 0; m < MM; m++) acc[m] += v * sQm[m * CIN + c];
    }
    for (int m = 0; m < MM; m++) {
      float s = acc[m] * invs;
      scx[(size_t)m * HW + l] = s;
      lmax[m] = fmaxf(lmax[m], s);
    }
  }
  for (int m = 0; m < MM; m++) {
    __syncthreads(); sRed[tid] = lmax[m]; __syncthreads();
    for (int s = 128; s > 0; s >>= 1) {
      if (tid < s) sRed[tid] = fmaxf(sRed[tid], sRed[tid + s]);
      __syncthreads();
    }
    if (tid == 0) sScal[m] = sRed[0];
  }
  __syncthreads();

  // exp + sum (store exp in-place)
  float lsum[MM] = {0, 0, 0, 0, 0, 0};
  for (int l = tid; l < HW; l += 256)
    for (int m = 0; m < MM; m++) {
      float e = __expf(scx[(size_t)m * HW + l] - sScal[m]);
      scx[(size_t)m * HW + l] = e;
      lsum[m] += e;
    }
  for (int m = 0; m < MM; m++) {
    __syncthreads(); sRed[tid] = lsum[m]; __syncthreads();
    for (int s = 128; s > 0; s >>= 1) {
      if (tid < s) sRed[tid] += sRed[tid + s];
      __syncthreads();
    }
    if (tid == 0) sScal[8 + m] = sRed[0];
  }
  __syncthreads();

  // fusion[m][c] = sum_l p * lf
  PFOR(i, MM * CIN) {
    int m = i >> 6, c = i & 63; float a = 0.f;
    const float* sp = scx + (size_t)m * HW;
    const float* xp = lf + (size_t)c * HW;
    for (int l = 0; l < HW; l++) a += sp[l] * xp[l];
    sFus[i] = a / sScal[8 + m];
  }
  __syncthreads();

  // z_hidden = fusion @ mf_un_w + b + global
  {
    const float* W = P.p[P_MF_UN_W]; const float* b = P.p[P_MF_UN_B];
    PFOR(i, MM * DD) {
      int m = i / DD, d = i % DD; float a = b[d] + sG[i];
      for (int c = 0; c < CIN; c++) a += sFus[m * CIN + c] * W[c * DD + d];
      sZ[i] = a;
    }
  }
  __syncthreads();

  // qvk = relu6(z_hidden @ f_qvk_w + b)  (6x576)
  {
    const float* W = P.p[P_F_QVK_W]; const float* b = P.p[P_F_QVK_B];
    PFOR(i, MM * 576) {
      int m = i / 576, j = i % 576; float a = b[j];
      for (int d = 0; d < DD; d++) a += sZ[m * DD + d] * W[d * 576 + j];
      sA[i] = relu6f(a);
    }
  }
  __syncthreads();

  // Q, K, V = relu6(qvk_part @ W + b)  (branch select, no LDS pointer arrays)
  PFOR(i, 3 * MM * DD) {
    int which = i / (MM * DD); int r = i % (MM * DD);
    int m = r / DD, j = r % DD;
    const float* ww = (which == 0) ? P.p[P_F_Q_W]
                     : (which == 1) ? P.p[P_F_K_W] : P.p[P_F_V_W];
    const float* bb = (which == 0) ? P.p[P_F_Q_B]
                     : (which == 1) ? P.p[P_F_K_B] : P.p[P_F_V_B];
    float a = bb[j];
    const float* src = &sA[m * 576 + which * DD];
    for (int d = 0; d < DD; d++) a += src[d] * ww[d * DD + j];
    float vv = relu6f(a);
    if (which == 0)      sQh[r] = vv;
    else if (which == 1) sKh[r] = vv;
    else                 sVh[r] = vv;
  }
  __syncthreads();

  // head attention: sc[h][q][k], dph=24, HEAD=8, flat (m, d*8+h)
  const float invh = 1.0f / (sqrtf(24.0f) + 1e-10f);
  PFOR(i, 8 * MM * MM) {
    int h = i / 36, r = i % 36, qi = r / 6, ki = r % 6;
    float a = 0.f;
    for (int d = 0; d < 24; d++) a += sQh[qi * DD + d * 8 + h] * sKh[ki * DD + d * 8 + h];
    sB[i] = a * invh;
  }
  __syncthreads();
  PFOR(i, 48) {  // softmax over k per (h,q)
    int base = i * 6;
    float mx = -1e30f;
    for (int k = 0; k < 6; k++) mx = fmaxf(mx, sB[base + k]);
    float s = 0.f;
    for (int k = 0; k < 6; k++) { float e = __expf(sB[base + k] - mx); sB[base + k] = e; s += e; }
    float r = 1.f / s;
    for (int k = 0; k < 6; k++) sB[base + k] *= r;
  }
  __syncthreads();
  // att_in[q][h*24+d] = sum_k V[k,d,h]*sm[h,q,k]
  PFOR(i, MM * DD) {
    int q = i / DD, j = i % DD, h = j / 24, d = j % 24;
    float a = 0.f;
    for (int k = 0; k < 6; k++) a += sVh[k * DD + d * 8 + h] * sB[(h * 6 + q) * 6 + k];
    sC[i] = a;
  }
  __syncthreads();
  // att = relu6(att_in @ f_o_w + b) -> sQh
  {
    const float* W = P.p[P_F_O_W]; const float* b = P.p[P_F_O_B];
    PFOR(i, MM * DD) {
      int m = i / DD, j = i % DD; float a = b[j];
      for (int d = 0; d < DD; d++) a += sC[m * DD + d] * W[d * DD + j];
      sQh[i] = relu6f(a);
    }
  }
  __syncthreads();
  // LN -> sKh
  if (tid < MM) {
    float mu = 0.f, ss = 0.f;
    for (int d = 0; d < DD; d++) { float x = sQh[tid * DD + d]; mu += x; ss += x * x; }
    mu /= (float)DD; ss = ss / (float)DD - mu * mu;
    sScal[16 + tid] = mu; sScal[24 + tid] = rsqrtf(ss + 1e-5f);
  }
  __syncthreads();
  {
    const float* g = P.p[P_F_NORM_G]; const float* b = P.p[P_F_NORM_B];
    PFOR(i, MM * DD) {
      int m = i / DD, j = i % DD;
      sKh[i] = (sQh[i] - sScal[16 + m]) * sScal[24 + m] * g[j] + b[j];
    }
  }
  __syncthreads();
  // mlp1 -> sB (6x384)
  {
    const float* W = P.p[P_F_MLP1_W]; const float* b = P.p[P_F_MLP1_B];
    PFOR(i, MM * 384) {
      int m = i / 384, j = i % 384; float a = b[j];
      for (int d = 0; d < DD; d++) a += sKh[m * DD + d] * W[d * 384 + j];
      sB[i] = relu6f(a);
    }
  }
  __syncthreads();
  // mlp2 -> sVh
  {
    const float* W = P.p[P_F_MLP2_W]; const float* b = P.p[P_F_MLP2_B];
    PFOR(i, MM * DD) {
      int m = i / DD, j = i % DD; float a = b[j];
      for (int d = 0; d < 384; d++) a += sB[m * 384 + d] * W[d * DD + j];
      sVh[i] = relu6f(a);
    }
  }
  __syncthreads();
  // LN -> z_out (sZ + global write)
  if (tid < MM) {
    float mu = 0.f, ss = 0.f;
    for (int d = 0; d < DD; d++) { float x = sVh[tid * DD + d]; mu += x; ss += x * x; }
    mu /= (float)DD; ss = ss / (float)DD - mu * mu;
    sScal[16 + tid] = mu; sScal[24 + tid] = rsqrtf(ss + 1e-5f);
  }
  __syncthreads();
  {
    const float* g = P.p[P_F_MLPNORM_G]; const float* b = P.p[P_F_MLPNORM_B];
    PFOR(i, MM * DD) {
      int m = i / DD, j = i % DD;
      float z = (sVh[i] - sScal[16 + m]) * sScal[24 + m] * g[j] + b[j];
      sZ[i] = z;
      zout[(size_t)n * MM * DD + i] = z;
    }
  }
  __syncthreads();

  // dyn-relu coefficients from ctrl = z_out[0,:] for ac1 and ac2
  for (int ac = 0; ac < 2; ac++) {
    const float* w1 = P.p[ac ? P_AC2_W1 : P_AC1_W1];
    const float* b1 = P.p[ac ? P_AC2_B1 : P_AC1_B1];
    const float* w2 = P.p[ac ? P_AC2_W2 : P_AC1_W2];
    const float* b2 = P.p[ac ? P_AC2_B2 : P_AC1_B2];
    const float* bg = P.p[ac ? P_AC2_BN_G : P_AC1_BN_G];
    const float* bb = P.p[ac ? P_AC2_BN_B : P_AC1_BN_B];
    const float* bm = P.p[ac ? P_AC2_BN_M : P_AC1_BN_M];
    const float* bv = P.p[ac ? P_AC2_BN_V : P_AC1_BN_V];
    float* Adst = ac ? A2 : A1;
    float* Bdst = ac ? B2c : B1c;
    PFOR(i, 768) {  // t1 = relu6(ctrl @ w1 + b1)
      float a = b1[i];
      for (int d = 0; d < DD; d++) a += sZ[d] * w1[d * 768 + i];
      sB[i] = relu6f(a);
    }
    __syncthreads();
    PFOR(i, 1024) {  // t2 -> relu6 -> bn -> 2*sigmoid-1
      float a = b2[i];
      for (int d = 0; d < 768; d++) a += sB[d] * w2[d * 1024 + i];
      a = relu6f(a);
      a = (a - bm[i]) * rsqrtf(bv[i] + 1e-5f) * bg[i] + bb[i];
      float th = 2.f / (1.f + __expf(-a)) - 1.f;
      if (i < 512) Adst[n * 512 + i] = th + (i < 256 ? 1.f : 0.f);
      else Bdst[n * 512 + i - 512] = 0.5f * th;
    }
    __syncthreads();
  }

  // kv = relu6(z_out @ fm_kv_w + b) -> Kf, Vf
  {
    const float* W = P.p[P_FM_KV_W]; const float* b = P.p[P_FM_KV_B];
    PFOR(i, MM * 128) {
      int m = i / 128, j = i % 128; float a = b[j];
      for (int d = 0; d < DD; d++) a += sZ[m * DD + d] * W[d * 128 + j];
      a = relu6f(a);
      if (j < 64) KF[(size_t)n * 384 + m * 64 + j] = a;
      else        VF[(size_t)n * 384 + m * 64 + (j - 64)] = a;
    }
  }
}

// ---------------------------------------------------------------------------
// K2/K4: 1x1 conv as GEMM on V_WMMA_F32_16X16X4_F32.
// Per wave: one 16(oc) x 16(l) output tile, IC/4 WMMA steps.
// A 16x4 f32 layout: lanes 0-15 -> M=lane, VGPR0 K=(half?2:0), VGPR1 K=+1.
// B 4x16 f32 mirrored with N across lanes. C/D: VGPR r -> M=r+(half?8:0), N=lane&15.
// Grid sized exactly so EXEC is all-ones (WMMA requirement).
// ---------------------------------------------------------------------------
template <int IC, int OC, bool DYN>
__global__ __launch_bounds__(256) void conv1x1_wmma_kernel(
    const float* __restrict__ X, const float* __restrict__ W,
    const float* __restrict__ cb, const float* __restrict__ g,
    const float* __restrict__ b, const float* __restrict__ bm,
    const float* __restrict__ bv,
    const float* __restrict__ Ac, const float* __restrict__ Bc,
    float* __restrict__ Y)
{
  const int lane = threadIdx.x & 31;
  const int wave = threadIdx.x >> 5;
  const int OCT = OC / 16;
  int tile = blockIdx.x * 8 + wave;               // 0 .. NB*OCT*196-1
  int n = tile / (OCT * 196);
  int r = tile % (OCT * 196);
  int oc0 = (r / 196) * 16;
  int l0 = (r % 196) * 16;

  const int half = lane >> 4, lr = lane & 15;
  const int ka = half * 2;

  const float* Wrow = W + (size_t)(oc0 + lr) * IC;
  const float* Bcol = X + (size_t)n * IC * HW + l0 + lr;

  v8f acc = {0.f, 0.f, 0.f, 0.f, 0.f, 0.f, 0.f, 0.f};
#pragma unroll 4
  for (int k0 = 0; k0 < IC; k0 += 4) {
    v2f af, bf;
    af.x = Wrow[k0 + ka];
    af.y = Wrow[k0 + ka + 1];
    bf.x = Bcol[(size_t)(k0 + ka) * HW];
    bf.y = Bcol[(size_t)(k0 + ka + 1) * HW];
    acc = __builtin_amdgcn_wmma_f32_16x16x4_f32(false, af, false, bf,
                                                (short)0, acc, false, false);
  }

  float* Yn = Y + ((size_t)n * OC + oc0) * HW + l0 + lr;
#pragma unroll
  for (int rr = 0; rr < 8; rr++) {
    int mrow = rr + half * 8;
    int oc = oc0 + mrow;
    float sc = g[oc] * rsqrtf(bv[oc] + 1e-5f);
    float sh = (cb[oc] - bm[oc]) * sc + b[oc];
    float val = acc[rr] * sc + sh;
    if constexpr (DYN) {
      int nb = n * 512;
      val = fmaxf(val * Ac[nb + oc] + Bc[nb + oc],
                  val * Ac[nb + 256 + oc] + Bc[nb + 256 + oc]);
    }
    Yn[(size_t)mrow * HW] = val;
  }
}

// ---------------------------------------------------------------------------
// K3: depthwise 3x3 conv (pad 1) + BN2 + dyn-relu(ac2). Thread per element.
// ---------------------------------------------------------------------------
__global__ __launch_bounds__(256) void dw_conv3x3_kernel(
    Ptrs P, const float* __restrict__ X,
    const float* __restrict__ Ac, const float* __restrict__ Bc,
    float* __restrict__ Y)
{
  size_t idx = (size_t)blockIdx.x * 256 + threadIdx.x;  // < 32*256*3136
  int l = (int)(idx % HW);
  int c = (int)((idx / HW) % CEXP);
  int n = (int)(idx / ((size_t)HW * CEXP));
  int y = l / 56, x = l % 56;
  const float* W = P.p[P_M_CONV2_W] + c * 9;
  const float* Xc = X + ((size_t)n * CEXP + c) * HW;
  float acc = P.p[P_M_CONV2_B][c];
#pragma unroll
  for (int ky = 0; ky < 3; ky++) {
    int yy = y + ky - 1;
    if (yy < 0 || yy >= 56) continue;
#pragma unroll
    for (int kx = 0; kx < 3; kx++) {
      int xx = x + kx - 1;
      if (xx < 0 || xx >= 56) continue;
      acc += Xc[yy * 56 + xx] * W[ky * 3 + kx];
    }
  }
  float sc = P.p[P_M_BN2_G][c] * rsqrtf(P.p[P_M_BN2_V][c] + 1e-5f);
  float v = (acc - P.p[P_M_BN2_M][c]) * sc + P.p[P_M_BN2_B][c];
  int nb = n * 512;
  Y[idx] = fmaxf(v * Ac[nb + c] + Bc[nb + c],
                 v * Ac[nb + 256 + c] + Bc[nb + 256 + c]);
}

// ---------------------------------------------------------------------------
// K5: Former->Mobile attention (k=6) + residual. Thread per pixel.
// ---------------------------------------------------------------------------
__global__ __launch_bounds__(256) void fm_attn_kernel(
    const float* __restrict__ xh, const float* __restrict__ KF,
    const float* __restrict__ VF, float* __restrict__ out)
{
  int n = blockIdx.y, tid = threadIdx.x;
  __shared__ float sK[MM * COUT], sV[MM * COUT];
  for (int i = tid; i < MM * COUT; i += 256) {
    sK[i] = KF[(size_t)n * MM * COUT + i];
    sV[i] = VF[(size_t)n * MM * COUT + i];
  }
  __syncthreads();
  int l = blockIdx.x * 256 + tid;
  if (l >= HW) return;
  const float* x = xh + (size_t)n * COUT * HW + l;
  float s[MM] = {0, 0, 0, 0, 0, 0};
  for (int c = 0; c < COUT; c++) {
    float v = x[(size_t)c * HW];
    for (int k = 0; k < MM; k++) s[k] += v * sK[k * COUT + c];
  }
  float mx = s[0];
  for (int k = 1; k < MM; k++) mx = fmaxf(mx, s[k]);
  float sum = 0.f;
  for (int k = 0; k < MM; k++) { s[k] = __expf(s[k] - mx); sum += s[k]; }
  float inv = 1.f / sum;
  for (int k = 0; k < MM; k++) s[k] *= inv;
  float* o = out + (size_t)n * COUT * HW + l;
  for (int c = 0; c < COUT; c++) {
    float v = x[(size_t)c * HW];
    float a = v;
    for (int k = 0; k < MM; k++) a += s[k] * sV[k * COUT + c];
    o[(size_t)c * HW] = a;
  }
}

// ---------------------------------------------------------------------------
// Launch. Workspace layout (floats):
//   buf1 [32*256*3136]  x1, later reused for x_hidden (first 32*64*3136)
//   buf2 [32*256*3136]  x2
//   scores [32*6*3136], a1/b1/a2/b2 [32*512 each], Kf/Vf [32*6*64 each]
// Total ~199 MiB.
// ---------------------------------------------------------------------------
extern "C" void kernel_launch(void* const* d_in, const int* in_sizes, int n_in,
                              void* d_out, int out_size, void* d_ws, size_t ws_size,
                              hipStream_t stream)
{
  (void)in_sizes; (void)out_size; (void)ws_size;
  Ptrs P;
  for (int i = 0; i < NPTRS; i++) P.p[i] = (i < n_in) ? (const float*)d_in[i] : nullptr;

  float* w = (float*)d_ws;
  const size_t S1 = (size_t)NB * CEXP * HW;  // 25,690,112
  float* buf1 = w;
  float* buf2 = w + S1;
  float* scores = w + 2 * S1;                 // 602,112
  float* A1 = scores + (size_t)NB * MM * HW;
  float* B1c = A1 + NB * 512;
  float* A2 = B1c + NB * 512;
  float* B2c = A2 + NB * 512;
  float* KF = B2c + NB * 512;
  float* VF = KF + NB * MM * COUT;

  float* xout = (float*)d_out;
  float* zout = xout + (size_t)NB * COUT * HW;  // x_out then z_out

  // K1: Former path (32 images)
  former_kernel<<<NB, 256, 0, stream>>>(P, scores, A1, B1c, A2, B2c, KF, VF, zout);

  // K2: conv1 (64->256) + bn1 + dyn-relu1 via WMMA; 32*16*196 tiles / 8 waves
  conv1x1_wmma_kernel<CIN, CEXP, true><<<(NB * (CEXP / 16) * 196) / 8, 256, 0, stream>>>(
      P.p[IN_LOCAL], P.p[P_M_CONV1_W], P.p[P_M_CONV1_B],
      P.p[P_M_BN1_G], P.p[P_M_BN1_B], P.p[P_M_BN1_M], P.p[P_M_BN1_V],
      A1, B1c, buf1);

  // K3: depthwise 3x3 + bn2 + dyn-relu2
  dw_conv3x3_kernel<<<(unsigned)(((size_t)NB * CEXP * HW) / 256), 256, 0, stream>>>(
      P, buf1, A2, B2c, buf2);

  // K4: conv3 (256->64) + bn3 via WMMA -> x_hidden (reuse buf1 region)
  conv1x1_wmma_kernel<CEXP, COUT, false><<<(NB * (COUT / 16) * 196) / 8, 256, 0, stream>>>(
      buf2, P.p[P_M_CONV3_W], P.p[P_M_CONV3_B],
      P.p[P_M_BN3_G], P.p[P_M_BN3_B], P.p[P_M_BN3_M], P.p[P_M_BN3_V],
      nullptr, nullptr, buf1);

  // K5: Former->Mobile attention + residual -> x_out
  fm_attn_kernel<<<dim3((HW + 255) / 256, NB), 256, 0, stream>>>(buf1, KF, VF, xout);
}